// LDS_41077067219061
// MI455X (gfx1250) — compile-verified
//
#include <hip/hip_runtime.h>

typedef float v2f __attribute__((ext_vector_type(2)));
typedef float v8f __attribute__((ext_vector_type(8)));

#define B_SZ      8
#define SEQ_LEN   2048
#define STATE_DIM 1024
#define D_IN      512
#define D_OUT     512
#define KX        10
#define M_ROWS    (B_SZ * SEQ_LEN)   // 16384 flattened (b,l) rows

// fp32 WMMA: D(16x16,f32) = A(16x4,f32) x B(4x16,f32) + C
// 8-arg form: (neg_a, A, neg_b, B, c_mod, C, reuse_a, reuse_b)
__device__ __forceinline__ v8f wmma_f32(v2f a, v2f b, v8f c) {
    return __builtin_amdgcn_wmma_f32_16x16x4_f32(
        false, a, false, b, (short)0, c, false, false);
}

// ---------------------------------------------------------------------------
// Repack M[o, i, k] (strides 5120,10,1) -> Mt[k][i][o] so conv B-fragments
// are contiguous along the N (=o) dimension.
// ---------------------------------------------------------------------------
__global__ void k_repackM(const float* __restrict__ M, float* __restrict__ Mt) {
    int idx = blockIdx.x * blockDim.x + threadIdx.x;
    if (idx >= KX * D_IN * D_OUT) return;
    int o = idx % D_OUT;
    int k = (idx / D_OUT) % D_IN;
    int i = idx / (D_OUT * D_IN);
    Mt[idx] = M[(size_t)o * (D_IN * KX) + (size_t)k * KX + i];
}

// ---------------------------------------------------------------------------
// uB[r, s] = X[r, :] @ B[:, s]   (r = 16384, K = 512, N = 1024)
// One wave -> 16x64 tile (4 fp32 WMMA accumulators); 8 waves -> 128x64/block.
// ---------------------------------------------------------------------------
__global__ __launch_bounds__(256) void k_proj(const float* __restrict__ X,
                                              const float* __restrict__ Bm,
                                              float* __restrict__ uB) {
    const int lane = threadIdx.x & 31;
    const int wave = threadIdx.x >> 5;
    const int m0   = blockIdx.x * 128 + wave * 16;
    const int n0   = blockIdx.y * 64;
    const int m    = lane & 15;          // A-frag row
    const int half = lane >> 4;
    const int kb   = half << 1;          // A/B frag k-offset within 4-step
    const int n    = n0 + (lane & 15);   // B/D frag column

    v8f acc[4] = {};
    const float* xrow = X + (size_t)(m0 + m) * D_IN;

    for (int k0 = 0; k0 < D_IN; k0 += 4) {
        v2f a = *(const v2f*)(xrow + k0 + kb);
#pragma unroll
        for (int c = 0; c < 4; ++c) {
            v2f b;
            b.x = Bm[(size_t)(k0 + kb)     * STATE_DIM + n + c * 16];
            b.y = Bm[(size_t)(k0 + kb + 1) * STATE_DIM + n + c * 16];
            acc[c] = wmma_f32(a, b, acc[c]);
        }
    }
#pragma unroll
    for (int c = 0; c < 4; ++c)
#pragma unroll
        for (int r = 0; r < 8; ++r)
            uB[(size_t)(m0 + half * 8 + r) * STATE_DIM + n + c * 16] = acc[c][r];
}

// ---------------------------------------------------------------------------
// In-place diagonal scan over uB: h_t = A[s]*h_{t-1} + uB[b,t,s]; uB <- h_t.
// One thread per (b, s) channel; consecutive threads -> consecutive s.
// ---------------------------------------------------------------------------
__global__ void k_scan(float* __restrict__ uB,
                       const float* __restrict__ A,
                       const float* __restrict__ h0) {
    int g = blockIdx.x * blockDim.x + threadIdx.x;   // 0 .. 8191
    int s = g & (STATE_DIM - 1);
    int b = g >> 10;
    float a = A[s];
    float h = h0[s];
    float* p = uB + (size_t)b * SEQ_LEN * STATE_DIM + s;
    for (int t = 0; t < SEQ_LEN; ++t) {
        float u = p[(size_t)t * STATE_DIM];
        h = fmaf(a, h, u);
        p[(size_t)t * STATE_DIM] = h;
    }
}

// ---------------------------------------------------------------------------
// out[r, o] = hs[r, :] @ C[:, o]  +  sum_{i<KX, i<=l} X[r-i, :] @ Mt[i][:, o]
// Fused: both reductions accumulate into the same WMMA accumulators.
// ---------------------------------------------------------------------------
__global__ __launch_bounds__(256) void k_out(const float* __restrict__ hs,
                                             const float* __restrict__ Cm,
                                             const float* __restrict__ X,
                                             const float* __restrict__ Mt,
                                             float* __restrict__ out) {
    const int lane = threadIdx.x & 31;
    const int wave = threadIdx.x >> 5;
    const int m0   = blockIdx.x * 128 + wave * 16;
    const int n0   = blockIdx.y * 64;
    const int m    = lane & 15;
    const int half = lane >> 4;
    const int kb   = half << 1;
    const int n    = n0 + (lane & 15);

    v8f acc[4] = {};

    // ---- hs @ C (K = 1024) ----
    const float* hrow = hs + (size_t)(m0 + m) * STATE_DIM;
    for (int k0 = 0; k0 < STATE_DIM; k0 += 4) {
        v2f a = *(const v2f*)(hrow + k0 + kb);
#pragma unroll
        for (int c = 0; c < 4; ++c) {
            v2f b;
            b.x = Cm[(size_t)(k0 + kb)     * D_OUT + n + c * 16];
            b.y = Cm[(size_t)(k0 + kb + 1) * D_OUT + n + c * 16];
            acc[c] = wmma_f32(a, b, acc[c]);
        }
    }

    // ---- 10-tap causal conv as shifted GEMMs (K = 512 per tap) ----
    const int l_local = (m0 + m) & (SEQ_LEN - 1);   // 16-row tiles never cross b
    for (int i = 0; i < KX; ++i) {
        const float* w     = Mt + (size_t)i * D_IN * D_OUT;
        const bool   valid = (l_local >= i);
        const float* xrow  = X + ((size_t)(m0 + m) - (size_t)i) * D_IN;
        for (int k0 = 0; k0 < D_IN; k0 += 4) {
            v2f a;
            if (valid) {
                a = *(const v2f*)(xrow + k0 + kb);
            } else {
                a.x = 0.0f; a.y = 0.0f;             // zero-fill, EXEC stays all-1s
            }
#pragma unroll
            for (int c = 0; c < 4; ++c) {
                v2f b;
                b.x = w[(size_t)(k0 + kb)     * D_OUT + n + c * 16];
                b.y = w[(size_t)(k0 + kb + 1) * D_OUT + n + c * 16];
                acc[c] = wmma_f32(a, b, acc[c]);
            }
        }
    }

#pragma unroll
    for (int c = 0; c < 4; ++c)
#pragma unroll
        for (int r = 0; r < 8; ++r)
            out[(size_t)(m0 + half * 8 + r) * D_OUT + n + c * 16] = acc[c][r];
}

// ---------------------------------------------------------------------------
extern "C" void kernel_launch(void* const* d_in, const int* in_sizes, int n_in,
                              void* d_out, int out_size, void* d_ws, size_t ws_size,
                              hipStream_t stream) {
    const float* X  = (const float*)d_in[0];   // [8, 2048, 512]
    const float* h0 = (const float*)d_in[1];   // [1024]
    const float* A  = (const float*)d_in[2];   // [1024]
    const float* Bm = (const float*)d_in[3];   // [512, 1024]
    const float* Cm = (const float*)d_in[4];   // [1024, 512]
    const float* M  = (const float*)d_in[5];   // [512, 512, 10]
    float* out = (float*)d_out;                // [8, 2048, 512]

    float* uB = (float*)d_ws;                            // 16384*1024 f32 (67 MB)
    float* Mt = uB + (size_t)M_ROWS * STATE_DIM;         // 10*512*512 f32 (10 MB)

    // 1) repack conv weights
    const int nM = KX * D_IN * D_OUT;
    k_repackM<<<(nM + 255) / 256, 256, 0, stream>>>(M, Mt);

    // 2) uB = X @ B (fp32 WMMA)
    dim3 g1(M_ROWS / 128, STATE_DIM / 64);
    k_proj<<<g1, 256, 0, stream>>>(X, Bm, uB);

    // 3) in-place diagonal scan (uB becomes hs)
    k_scan<<<(B_SZ * STATE_DIM) / 256, 256, 0, stream>>>(uB, A, h0);

    // 4) out = hs @ C + 10-tap causal conv (fused fp32 WMMA)
    dim3 g3(M_ROWS / 128, D_OUT / 64);
    k_out<<<g3, 256, 0, stream>>>(uB, Cm, X, Mt, out);
}